// EnhancedSpatialInSARModel_85779086835994
// MI455X (gfx1250) — compile-verified
//
#include <hip/hip_runtime.h>
#include <hip/hip_bf16.h>
#include <math.h>

// ---------------------------------------------------------------------------
// signals[n,t] = c[n] + l[n]*t + sum_i amp_i[n]*sin(w_i*t + ph_i[n])
//             = Coeff(N x 16) @ Basis(16 x Tpad)   (rank-10, padded to 16)
// Coeff row: [c, l, A0*cos(p0), A0*sin(p0), ..., A3*cos(p3), A3*sin(p3), 0...]
// Basis col: [1, t, sin(w0 t), cos(w0 t), ..., sin(w3 t), cos(w3 t), 0...]
// ---------------------------------------------------------------------------

#define N_ST     50000
#define T_PTS    1000
#define TPAD     1008          // 63 * 16
#define NT_TILES 63
#define NTILES_N 3125          // 50000 / 16
#define TWO_PI_F 6.28318530717958647692f

typedef __attribute__((ext_vector_type(2))) float v2f;
typedef __attribute__((ext_vector_type(8))) float v8f;

// ws layout (floats):
//   [0..64]    cluster accumulators: per cluster c (13 each):
//                [c*13+i]    sum amp_i            (i=0..3)
//                [c*13+4+i]  sum cos(ph_i)
//                [c*13+8+i]  sum sin(ph_i)
//                [c*13+12]   count
//   [128..167] cluster means: [c*8+i]=meanAmp_i, [c*8+4+i]=meanPh_i
//   [168..172] counts[c]
//   [256..)    basis: 16 x TPAD row-major

// ------------------------- kernel 1: zero accumulators ----------------------
__global__ void k_zero_acc(float* acc) {
    int t = threadIdx.x;
    if (t < 65) acc[t] = 0.0f;
}

// ------------------------- kernel 2: cluster accumulate ---------------------
__global__ void k_cluster_accum(const float* __restrict__ amps,
                                const float* __restrict__ phs,
                                const int*   __restrict__ labels,
                                float* __restrict__ acc) {
    __shared__ float s[65];
    int tid = threadIdx.x;
    if (tid < 65) s[tid] = 0.0f;
    __syncthreads();
    int n = blockIdx.x * blockDim.x + tid;
    if (n < N_ST) {
        int lab = labels[n];
        atomicAdd(&s[lab * 13 + 12], 1.0f);
        #pragma unroll
        for (int i = 0; i < 4; ++i) {
            float a = amps[n * 4 + i];
            float p = phs[n * 4 + i];
            atomicAdd(&s[lab * 13 + i],     a);
            atomicAdd(&s[lab * 13 + 4 + i], cosf(p));
            atomicAdd(&s[lab * 13 + 8 + i], sinf(p));
        }
    }
    __syncthreads();
    if (tid < 65) atomicAdd(&acc[tid], s[tid]);
}

// ------------------------- kernel 3: cluster finalize -----------------------
__global__ void k_cluster_final(const float* __restrict__ acc,
                                float* __restrict__ meanv,
                                float* __restrict__ counts) {
    int c = threadIdx.x;
    if (c < 5) {
        float cnt = acc[c * 13 + 12];
        counts[c] = cnt;
        float inv = 1.0f / fmaxf(cnt, 1.0f);
        #pragma unroll
        for (int i = 0; i < 4; ++i) {
            meanv[c * 8 + i]     = acc[c * 13 + i] * inv;
            meanv[c * 8 + 4 + i] = atan2f(acc[c * 13 + 8 + i], acc[c * 13 + 4 + i]);
        }
    }
}

// ------------------------- kernel 4: basis table ----------------------------
__global__ void k_basis(const float* __restrict__ timev, float* __restrict__ basis) {
    int t = blockIdx.x * blockDim.x + threadIdx.x;
    if (t >= TPAD) return;
    float tv = (t < T_PTS) ? timev[t] : 0.0f;
    basis[0 * TPAD + t] = 1.0f;
    basis[1 * TPAD + t] = tv;
    const float freqs[4] = {4.0f, 2.0f, 1.0f, 0.5f};   // 1/period
    #pragma unroll
    for (int i = 0; i < 4; ++i) {
        float w = TWO_PI_F * freqs[i];
        basis[(2 + 2 * i) * TPAD + t] = sinf(w * tv);
        basis[(3 + 2 * i) * TPAD + t] = cosf(w * tv);
    }
    #pragma unroll
    for (int k = 10; k < 16; ++k) basis[k * TPAD + t] = 0.0f;
}

// ------------------- kernel 5: message passing + WMMA GEMM ------------------
// One block (256 thr = 8 waves) per 16-station tile. Threads 0..63 compute the
// message-passed coefficients into LDS, then each wave sweeps T-tiles with
// V_WMMA_F32_16X16X4_F32 (K=16 as 4 chained K=4 steps).
__global__ void __launch_bounds__(256)
k_mp_gemm(const float* __restrict__ coff,    // constant_offset
          const float* __restrict__ ltr,     // linear_trend
          const float* __restrict__ amps,    // (N,4)
          const float* __restrict__ phs,     // (N,4)
          const float* __restrict__ lw,      // (N,5)
          const float* __restrict__ rw,      // (N,15)
          const int*   __restrict__ lidx,    // (N,5)
          const int*   __restrict__ ridx,    // (N,15)
          const int*   __restrict__ labels,  // (N)
          const float* __restrict__ meanv,   // (5,8)
          const float* __restrict__ counts,  // (5)
          const float* __restrict__ basis,   // (16,TPAD)
          float* __restrict__ out) {         // (N,T)
    __shared__ float cLDS[16][17];           // 17 stride: bank-conflict-free
    const int tid = threadIdx.x;
    const int rowBase = blockIdx.x * 16;

    // ---- phase 1: build coefficient tile ----
    if (tid < 64) {
        int m = tid >> 2, i = tid & 3;
        int n = rowBase + m;
        int lab = labels[n];
        float cnt = counts[lab];

        // amplitude message passing
        float a0 = amps[n * 4 + i];
        float lu = 0.0f;
        #pragma unroll
        for (int k = 0; k < 5; ++k) {
            int j = lidx[n * 5 + k];
            lu += amps[j * 4 + i] * lw[n * 5 + k];
        }
        float ru = 0.0f;
        #pragma unroll
        for (int k = 0; k < 15; ++k) {
            int j = ridx[n * 15 + k];
            ru += amps[j * 4 + i] * rw[n * 15 + k];
        }
        ru *= 0.7f;
        float cu = (cnt > 1.0f) ? meanv[lab * 8 + i] : a0;
        float ampv = 0.7f * a0 + 0.3f * (0.5f * lu + 0.3f * ru + 0.2f * cu);

        // phase message passing (circular means)
        float p0 = phs[n * 4 + i];
        float lc = 0.0f, ls = 0.0f;
        #pragma unroll
        for (int k = 0; k < 5; ++k) {
            int j = lidx[n * 5 + k];
            float ph = phs[j * 4 + i], w = lw[n * 5 + k];
            lc += cosf(ph) * w; ls += sinf(ph) * w;
        }
        float lup = atan2f(ls, lc);
        float rc = 0.0f, rs = 0.0f;
        #pragma unroll
        for (int k = 0; k < 15; ++k) {
            int j = ridx[n * 15 + k];
            float ph = phs[j * 4 + i], w = rw[n * 15 + k];
            rc += cosf(ph) * w; rs += sinf(ph) * w;
        }
        float rup = atan2f(rs, rc);
        float cup = (cnt > 1.0f) ? meanv[lab * 8 + 4 + i] : p0;
        float phm = 0.7f * p0 + 0.3f * (0.5f * lup + 0.3f * rup + 0.2f * cup);

        cLDS[m][2 + 2 * i] = ampv * cosf(phm);
        cLDS[m][3 + 2 * i] = ampv * sinf(phm);
    } else if (tid < 80) {
        int m = tid - 64;
        int n = rowBase + m;
        cLDS[m][0] = coff[n];
        cLDS[m][1] = ltr[n];
    } else if (tid < 176) {
        int u = tid - 80;          // 16 rows x 6 pad cols
        cLDS[u / 6][10 + u % 6] = 0.0f;
    }
    __syncthreads();

    // ---- phase 2: WMMA sweep over T tiles ----
    const int wave = tid >> 5;
    const int lane = tid & 31;
    const int M  = lane & 15;
    const int hi = lane >> 4;

    // A operands: lanes 0-15 hold K={4j,4j+1}, lanes 16-31 hold K={4j+2,4j+3}
    v2f a[4];
    #pragma unroll
    for (int j = 0; j < 4; ++j) {
        int kc = 4 * j + 2 * hi;
        a[j].x = cLDS[M][kc];
        a[j].y = cLDS[M][kc + 1];
    }

    for (int tIdx = wave; tIdx < NT_TILES; tIdx += 8) {
        const int colBase = tIdx * 16;
        const int cg = colBase + M;      // this lane's output column
        v8f acc = {};
        #pragma unroll
        for (int j = 0; j < 4; ++j) {
            int kc = 4 * j + 2 * hi;
            v2f b;
            b.x = basis[(size_t)kc * TPAD + cg];
            b.y = basis[(size_t)(kc + 1) * TPAD + cg];
            acc = __builtin_amdgcn_wmma_f32_16x16x4_f32(
                false, a[j], false, b, (short)0, acc, false, false);
        }
        if (cg < T_PTS) {
            float* outp = out + (size_t)(rowBase + hi * 8) * T_PTS + cg;
            #pragma unroll
            for (int r = 0; r < 8; ++r)
                __builtin_nontemporal_store(acc[r], outp + (size_t)r * T_PTS);
        }
    }
}

// ---------------------------------------------------------------------------
extern "C" void kernel_launch(void* const* d_in, const int* in_sizes, int n_in,
                              void* d_out, int out_size, void* d_ws, size_t ws_size,
                              hipStream_t stream) {
    const float* timev  = (const float*)d_in[0];
    const float* coff   = (const float*)d_in[1];
    const float* ltr    = (const float*)d_in[2];
    const float* amps   = (const float*)d_in[3];
    const float* phs    = (const float*)d_in[4];
    const float* lw     = (const float*)d_in[5];
    const float* rw     = (const float*)d_in[6];
    const int*   lidx   = (const int*)d_in[7];
    const int*   ridx   = (const int*)d_in[8];
    const int*   labels = (const int*)d_in[9];
    float* out = (float*)d_out;

    float* ws     = (float*)d_ws;
    float* acc    = ws;          // 65 floats
    float* meanv  = ws + 128;    // 40 floats
    float* counts = ws + 168;    // 5 floats
    float* basis  = ws + 256;    // 16*TPAD floats

    k_zero_acc<<<1, 128, 0, stream>>>(acc);
    k_cluster_accum<<<(N_ST + 255) / 256, 256, 0, stream>>>(amps, phs, labels, acc);
    k_cluster_final<<<1, 32, 0, stream>>>(acc, meanv, counts);
    k_basis<<<(TPAD + 255) / 256, 256, 0, stream>>>(timev, basis);
    k_mp_gemm<<<NTILES_N, 256, 0, stream>>>(coff, ltr, amps, phs, lw, rw,
                                            lidx, ridx, labels, meanv, counts,
                                            basis, out);
}